// GraphAttentionLayer_33251636806287
// MI455X (gfx1250) — compile-verified
//
#include <hip/hip_runtime.h>
#include <hip/hip_bf16.h>
#include <math.h>

// GAT layer, MI455X (gfx1250, wave32).
// Roofline: adj read (256 MB) dominates -> ~11.5us @ 23.3 TB/s. Single pass
// over adj with fused mask+softmax+AV using fp32 WMMA (16x16x4).

#define N_NODES 8192
#define DIN     512
#define DOUT    64
#define LALPHA  0.2f
#define WSTR    66      // padded LDS row stride (floats) to kill bank conflicts

typedef float v2f __attribute__((ext_vector_type(2)));
typedef float v8f __attribute__((ext_vector_type(8)));

__device__ __forceinline__ v8f wmma_f32(v2f a, v2f b, v8f c) {
  // D = A(16x4) * B(4x16) + C(16x16), all f32. 8-arg VOP3P form:
  // (neg_a, A, neg_b, B, c_mod, C, reuse_a, reuse_b)
  return __builtin_amdgcn_wmma_f32_16x16x4_f32(false, a, false, b, (short)0, c,
                                               false, false);
}

// ---------------------------------------------------------------------------
// Kernel 1: Wh = h @ W  (8192x512 * 512x64), fp32 WMMA 16x16x4.
// Each block: 16 rows x 64 cols, 4 waves (one 16-col tile per wave).
// Also computes f1 = Wh@a[:64], f2 = Wh@a[64:] per row from the LDS tile.
// ---------------------------------------------------------------------------
__global__ __launch_bounds__(128) void k_gemm_wh(
    const float* __restrict__ h, const float* __restrict__ W,
    const float* __restrict__ a, float* __restrict__ Wh,
    float* __restrict__ f1, float* __restrict__ f2) {
  __shared__ __align__(16) float tile[16 * WSTR];

  const int t    = threadIdx.x;
  const int lane = t & 31;
  const int wave = t >> 5;
  const int i0   = blockIdx.x * 16;
  const int col0 = wave * 16;
  const int rn   = lane & 15;        // A: row M, B: col N
  const int koff = (lane >> 4) * 2;  // K pair select

  v8f acc = {};
  const float* arow = h + (size_t)(i0 + rn) * DIN;
#pragma unroll 4
  for (int k = 0; k < DIN; k += 4) {
    v2f af = *(const v2f*)(arow + k + koff);               // h[i0+rn][k+koff .. +1]
    v2f bf;
    bf.x = W[(size_t)(k + koff) * DOUT + col0 + rn];       // W[k+koff][col]
    bf.y = W[(size_t)(k + koff + 1) * DOUT + col0 + rn];
    acc = wmma_f32(af, bf, acc);
  }

  // C layout: lane holds col = col0+rn, rows M = v + 8*(lane>>4), v=0..7
#pragma unroll
  for (int v = 0; v < 8; ++v) {
    int m = v + 8 * (lane >> 4);
    tile[m * WSTR + col0 + rn] = acc[v];
  }
  __syncthreads();

  // coalesced Wh writeback: 1024 floats, 128 threads x 8
#pragma unroll
  for (int q = 0; q < 8; ++q) {
    int idx = t + q * 128;
    Wh[(size_t)i0 * DOUT + idx] = tile[(idx >> 6) * WSTR + (idx & 63)];
  }

  // f1/f2: 16 threads, one row each, 64-length dots with a1/a2
  if (t < 16) {
    float s1 = 0.f, s2 = 0.f;
#pragma unroll
    for (int c = 0; c < DOUT; ++c) {
      float v = tile[t * WSTR + c];
      s1 += v * a[c];
      s2 += v * a[DOUT + c];
    }
    f1[i0 + t] = s1;
    f2[i0 + t] = s2;
  }
}

// ---------------------------------------------------------------------------
// Kernel 2: deterministic reductions: f2max = max_j f2[j];
// colsum[c] = sum_r Wh[r][c]  (exact fallback for a fully-masked row).
// Single block, 256 threads.
// ---------------------------------------------------------------------------
__global__ __launch_bounds__(256) void k_reduce(
    const float* __restrict__ f2, const float* __restrict__ Wh,
    float* __restrict__ f2mx, float* __restrict__ colsum) {
  __shared__ float smax[256];
  __shared__ float scol[4][64];
  const int t = threadIdx.x;

  float m = -3.0e38f;
  for (int i = t; i < N_NODES; i += 256) m = fmaxf(m, f2[i]);
  smax[t] = m;
  __syncthreads();
  for (int s = 128; s > 0; s >>= 1) {
    if (t < s) smax[t] = fmaxf(smax[t], smax[t + s]);
    __syncthreads();
  }
  if (t == 0) *f2mx = smax[0];

  const int c = t & 63, ch = t >> 6;
  float acc = 0.f;
  for (int r = ch; r < N_NODES; r += 4) acc += Wh[(size_t)r * DOUT + c];
  scol[ch][c] = acc;
  __syncthreads();
  if (t < 64) colsum[t] = scol[0][t] + scol[1][t] + scol[2][t] + scol[3][t];
}

// ---------------------------------------------------------------------------
// Kernel 3: fused mask + softmax + attention@Wh + ELU, single pass over adj.
// Block = 16 output rows, 128 threads (4 waves; wave w -> cols 16w..16w+15).
// Per 64-wide j-chunk: compute w[i][j] = adj ? exp(leakyrelu(f1+f2)-Mi) : 0
// into LDS, then 16 WMMA f32 16x16x4 steps per wave accumulate w @ Wh.
// Mi = leakyrelu(f1[i] + max f2) is a valid per-row softmax shift (monotone
// leakyrelu), so exp <= 1: no overflow, no per-row max pass needed.
// ---------------------------------------------------------------------------
__global__ __launch_bounds__(128) void k_attn(
    const int* __restrict__ adj, const float* __restrict__ Wh,
    const float* __restrict__ f1, const float* __restrict__ f2,
    const float* __restrict__ f2mxp, const float* __restrict__ colsum,
    float* __restrict__ out) {
  __shared__ __align__(16) float wlds[16 * WSTR];
  __shared__ float mrow[16];
  __shared__ float zred[128];
  __shared__ float zrow[16];

  const int t    = threadIdx.x;
  const int lane = t & 31;
  const int wave = t >> 5;
  const int i0   = blockIdx.x * 16;
  const int col0 = wave * 16;
  const int rn   = lane & 15;
  const int koff = (lane >> 4) * 2;

  const int r = t & 15;   // score row owned by this thread
  const int g = t >> 4;   // 8 groups x 8 j's cover a 64-wide chunk

  if (t < 16) {
    float x = f1[i0 + t] + *f2mxp;
    mrow[t] = x > 0.f ? x : LALPHA * x;
  }
  __syncthreads();

  const float f1r = f1[i0 + r];
  const float mr  = mrow[r];
  const int* arow = adj + (size_t)(i0 + r) * N_NODES + g * 8;

  v8f acc = {};
  float zpart = 0.f;

  for (int jb = 0; jb < N_NODES; jb += 64) {
    // prefetch next adj chunk into cache (global_prefetch_b8)
    if (jb + 64 < N_NODES) __builtin_prefetch(arow + jb + 64, 0, 1);

    // score+mask+exp for this 16x64 block
#pragma unroll
    for (int u = 0; u < 8; ++u) {
      int j  = jb + g * 8 + u;
      int av = arow[jb + u];
      float s = f1r + f2[j];
      s = s > 0.f ? s : LALPHA * s;
      float wv = (av > 0) ? __expf(s - mr) : 0.f;
      wlds[r * WSTR + g * 8 + u] = wv;
      zpart += wv;
    }
    __syncthreads();

    // acc += w(16x64) @ Wh(64x16-col-slice), 16 K=4 WMMA steps
#pragma unroll
    for (int ks = 0; ks < 64; ks += 4) {
      v2f af = *(const v2f*)(&wlds[rn * WSTR + ks + koff]);
      int jr = jb + ks + koff;
      v2f bf;
      bf.x = Wh[(size_t)jr * DOUT + col0 + rn];
      bf.y = Wh[(size_t)(jr + 1) * DOUT + col0 + rn];
      acc = wmma_f32(af, bf, acc);
    }
    __syncthreads();  // before wlds is overwritten
  }

  // reduce Z per row across the 8 groups
  zred[t] = zpart;
  __syncthreads();
  if (t < 16) {
    float z = 0.f;
#pragma unroll
    for (int gg = 0; gg < 8; ++gg) z += zred[t + 16 * gg];
    zrow[t] = z;
  }
  __syncthreads();

  // normalize + ELU + store. Lane holds col0+rn, rows m = v + 8*(lane>>4).
#pragma unroll
  for (int v = 0; v < 8; ++v) {
    int m   = v + 8 * (lane >> 4);
    float z = zrow[m];
    float val = (z > 0.f) ? acc[v] / z
                          : colsum[col0 + rn] * (1.0f / (float)N_NODES);
    val = val > 0.f ? val : (__expf(val) - 1.0f);   // elu
    out[(size_t)(i0 + m) * DOUT + col0 + rn] = val;
  }
}

// ---------------------------------------------------------------------------
extern "C" void kernel_launch(void* const* d_in, const int* in_sizes, int n_in,
                              void* d_out, int out_size, void* d_ws,
                              size_t ws_size, hipStream_t stream) {
  const float* h   = (const float*)d_in[0];  // [8192,512]
  const float* W   = (const float*)d_in[1];  // [512,64]
  const float* a   = (const float*)d_in[2];  // [128,1]
  const int*   adj = (const int*)d_in[3];    // [8192,8192]
  float*       out = (float*)d_out;          // [8192,64]

  char*  ws     = (char*)d_ws;
  float* Wh     = (float*)ws;                              // 2 MB
  float* f1     = (float*)(ws + (size_t)N_NODES * DOUT * 4);
  float* f2     = f1 + N_NODES;
  float* f2mx   = f2 + N_NODES;
  float* colsum = f2mx + 16;

  k_gemm_wh<<<N_NODES / 16, 128, 0, stream>>>(h, W, a, Wh, f1, f2);
  k_reduce<<<1, 256, 0, stream>>>(f2, Wh, f2mx, colsum);
  k_attn<<<N_NODES / 16, 128, 0, stream>>>(adj, Wh, f1, f2, f2mx, colsum, out);
}